// GraphConvolution_31756988187311
// MI455X (gfx1250) — compile-verified
//
#include <hip/hip_runtime.h>
#include <hip/hip_bf16.h>

#define N_NODES 100000
#define D_IN    256
#define D_OUT   128

typedef float v2f __attribute__((ext_vector_type(2)));
typedef float v8f __attribute__((ext_vector_type(8)));

// ---------------------------------------------------------------------------
// Kernel 1: zero the aggregation buffer in workspace (harness does not
// re-zero between replays).
// ---------------------------------------------------------------------------
__global__ void gc_zero_kernel(float4* __restrict__ agg4, int n4) {
    int i = blockIdx.x * blockDim.x + threadIdx.x;
    int stride = gridDim.x * blockDim.x;
    float4 z = make_float4(0.f, 0.f, 0.f, 0.f);
    for (; i < n4; i += stride) agg4[i] = z;
}

// ---------------------------------------------------------------------------
// Kernel 2: support = input @ W + bias  via V_WMMA_F32_16X16X4_F32.
// Block = 256 threads = 8 waves. Each wave owns a 16-row strip and all
// 128 output columns (8 accumulator tiles of 16x16), looping K in steps of 4.
//
// Fragment layouts (ISA 7.12.2, 32-bit):
//   A 16x4 : lane%16 -> M, (lane/16)*2 + vgpr -> K        (2 VGPRs)
//   B 4x16 : lane%16 -> N, (lane/16)*2 + vgpr -> K        (2 VGPRs, col-major)
//   C 16x16: lane%16 -> N, vgpr + 8*(lane/16) -> M        (8 VGPRs)
// ---------------------------------------------------------------------------
__global__ void __launch_bounds__(256)
gc_gemm_kernel(const float* __restrict__ A,    // [N_NODES, D_IN]
               const float* __restrict__ W,    // [D_IN, D_OUT]
               const float* __restrict__ bias, // [D_OUT]
               float* __restrict__ S)          // [N_NODES, D_OUT]
{
    const int lane  = threadIdx.x & 31;
    const int wave  = threadIdx.x >> 5;                 // 0..7
    const int mtile = blockIdx.x * 8 + wave;            // 16-row tile index
    const int row0  = mtile * 16;
    if (row0 >= N_NODES) return;                        // wave-uniform: EXEC stays all-1s

    const int m     = lane & 15;                        // M (for A) / N (for B,C)
    const int khalf = lane >> 4;                        // 0..1

    v8f acc[8];
#pragma unroll
    for (int t = 0; t < 8; ++t) acc[t] = (v8f)(0.0f);

    const float* __restrict__ arow = A + (size_t)(row0 + m) * D_IN;

    for (int k = 0; k < D_IN; k += 4) {
        const int ka = k + 2 * khalf;
        // A fragment: A[row0+m][ka], A[row0+m][ka+1]  (one b64 load)
        v2f afrag = *(const v2f*)(arow + ka);
#pragma unroll
        for (int t = 0; t < 8; ++t) {
            const int n = t * 16 + m;
            v2f bfrag;
            bfrag.x = W[(size_t)(ka + 0) * D_OUT + n];
            bfrag.y = W[(size_t)(ka + 1) * D_OUT + n];
            acc[t] = __builtin_amdgcn_wmma_f32_16x16x4_f32(
                /*neg_a=*/false, afrag, /*neg_b=*/false, bfrag,
                /*c_mod=*/(short)0, acc[t], /*reuse_a=*/false, /*reuse_b=*/false);
        }
    }

    // bias + store (C layout: VGPR i holds M = row0 + i + 8*khalf, N = t*16 + m)
#pragma unroll
    for (int t = 0; t < 8; ++t) {
        const int n  = t * 16 + m;
        const float bv = bias[n];
#pragma unroll
        for (int i = 0; i < 8; ++i) {
            const int r = row0 + i + 8 * khalf;
            S[(size_t)r * D_OUT + n] = acc[t][i] + bv;
        }
    }
}

// ---------------------------------------------------------------------------
// Kernel 3: edge aggregation. agg[row[e]] += val[e] * support[col[e]].
// One wave per edge; 32 edges' metadata loaded coalesced, then broadcast via
// __shfl (wave32). Per edge: one float4 (b128) gather per lane + 4 f32 atomic
// adds. support (51 MB) and agg (51 MB) are L2-resident on MI455X (192 MB L2),
// so both the gather and the atomics resolve in L2.
// ---------------------------------------------------------------------------
__global__ void __launch_bounds__(256)
gc_agg_kernel(const int*   __restrict__ erow,
              const int*   __restrict__ ecol,
              const float* __restrict__ eval,
              const float* __restrict__ S,     // [N_NODES, D_OUT]
              float*       __restrict__ agg,   // [N_NODES, D_OUT]
              int nEdges)
{
    const int lane   = threadIdx.x & 31;
    const int gwave  = (blockIdx.x * blockDim.x + threadIdx.x) >> 5;
    const int nwaves = (gridDim.x * blockDim.x) >> 5;

    for (int base = gwave * 32; base < nEdges; base += nwaves * 32) {
        const int e = base + lane;
        int r = 0, c = 0; float v = 0.0f;
        if (e < nEdges) { r = erow[e]; c = ecol[e]; v = eval[e]; }
        const int cnt = min(32, nEdges - base);
        for (int j = 0; j < cnt; ++j) {
            const int   rj = __shfl(r, j);
            const int   cj = __shfl(c, j);
            const float vj = __shfl(v, j);
            const float4 sv = *(const float4*)(S + (size_t)cj * D_OUT + lane * 4);
            float* dst = agg + (size_t)rj * D_OUT + lane * 4;
            atomicAdd(dst + 0, vj * sv.x);
            atomicAdd(dst + 1, vj * sv.y);
            atomicAdd(dst + 2, vj * sv.z);
            atomicAdd(dst + 3, vj * sv.w);
        }
    }
}

// ---------------------------------------------------------------------------
// Kernel 4: out = tanh(agg)
// ---------------------------------------------------------------------------
__global__ void gc_tanh_kernel(const float4* __restrict__ agg4,
                               float4* __restrict__ out4, int n4) {
    int i = blockIdx.x * blockDim.x + threadIdx.x;
    int stride = gridDim.x * blockDim.x;
    for (; i < n4; i += stride) {
        float4 a = agg4[i];
        float4 o;
        o.x = tanhf(a.x); o.y = tanhf(a.y); o.z = tanhf(a.z); o.w = tanhf(a.w);
        out4[i] = o;
    }
}

extern "C" void kernel_launch(void* const* d_in, const int* in_sizes, int n_in,
                              void* d_out, int out_size, void* d_ws, size_t ws_size,
                              hipStream_t stream) {
    const float* input    = (const float*)d_in[0];  // [N_NODES, D_IN]
    const int*   edge_row = (const int*)  d_in[1];  // [N_EDGES]
    const int*   edge_col = (const int*)  d_in[2];  // [N_EDGES]
    const float* edge_val = (const float*)d_in[3];  // [N_EDGES]
    const float* W        = (const float*)d_in[4];  // [D_IN, D_OUT]
    const float* bias     = (const float*)d_in[5];  // [D_OUT]
    float*       out      = (float*)d_out;          // [N_NODES, D_OUT]
    float*       agg      = (float*)d_ws;           // [N_NODES, D_OUT] scratch

    const int nEdges = in_sizes[1];
    const int nOut   = N_NODES * D_OUT;             // == out_size
    const int n4     = nOut / 4;

    // 1) zero agg in workspace
    {
        int blocks = (n4 + 255) / 256;
        if (blocks > 4096) blocks = 4096;
        gc_zero_kernel<<<blocks, 256, 0, stream>>>((float4*)agg, n4);
    }
    // 2) support = input @ W + b   (support lives in d_out for now)
    {
        const int mtiles = N_NODES / 16;            // 6250, exact
        const int blocks = (mtiles + 7) / 8;        // 782
        gc_gemm_kernel<<<blocks, 256, 0, stream>>>(input, W, bias, out);
    }
    // 3) agg += sparse adj @ support
    {
        gc_agg_kernel<<<1280, 256, 0, stream>>>(edge_row, edge_col, edge_val,
                                                out, agg, nEdges);
    }
    // 4) out = tanh(agg)
    {
        int blocks = (n4 + 255) / 256;
        if (blocks > 4096) blocks = 4096;
        gc_tanh_kernel<<<blocks, 256, 0, stream>>>((const float4*)agg,
                                                   (float4*)out, n4);
    }
}